// GraphConvRNN_71691594104970
// MI455X (gfx1250) — compile-verified
//
#include <hip/hip_runtime.h>
#include <hip/hip_bf16.h>
#include <math.h>

typedef __attribute__((ext_vector_type(2))) float v2f;
typedef __attribute__((ext_vector_type(8))) float v8f;

#define IN_C 128
#define HDIM 256
#define G3   768   // 3*H

// ------------------------------------------------------------------
// Zero a float region (grid-stride)
// ------------------------------------------------------------------
__global__ void k_zero(float* __restrict__ p, long n) {
  long i = (long)blockIdx.x * blockDim.x + threadIdx.x;
  long st = (long)gridDim.x * blockDim.x;
  for (; i < n; i += st) p[i] = 0.0f;
}

// ------------------------------------------------------------------
// WMMA f32 GEMM: C[M,N] = A[M,K] @ B (+bias)
// TRANSB=false: B is [K,N] row-major.  TRANSB=true: B is [N,K] (B^T).
// One wave32 computes a 16x32 strip: two 16x16 accumulators share one
// A fragment, K stepped by 4 via V_WMMA_F32_16X16X4_F32.
// ------------------------------------------------------------------
template <bool TRANSB, bool BIAS>
__global__ void k_gemm_wmma(const float* __restrict__ A, const float* __restrict__ B,
                            const float* __restrict__ bias, float* __restrict__ C,
                            int M, int N, int K) {
  const int wave = (int)((blockIdx.x * blockDim.x + threadIdx.x) >> 5);
  const int lane = threadIdx.x & 31;
  const int npairs = N >> 5;                 // 16x32 strips per row
  const int mt = wave / npairs;
  const int np = wave - mt * npairs;
  if (mt * 16 >= M) return;                  // uniform per wave
  const int m0 = mt << 4, n0 = np << 5;
  const int mrow  = m0 + (lane & 15);        // A-frag row (ISA 16x4 f32 layout)
  const int ncol0 = n0 + (lane & 15);        // B/C-frag col, tile 0
  const int ncol1 = ncol0 + 16;              // tile 1
  const int khalf = (lane >> 4) << 1;        // lanes 16-31 hold K+2,K+3

  const float* __restrict__ arow = A + (size_t)mrow * K;
  v8f acc0 = {}, acc1 = {};

#pragma unroll 4
  for (int k0 = 0; k0 < K; k0 += 4) {
    const int ka = k0 + khalf;
    v2f a, b0, b1;
    a.x = arow[ka];
    a.y = arow[ka + 1];
    if constexpr (TRANSB) {                  // b[v] = B[ncol][ka+v]
      b0.x = B[(size_t)ncol0 * K + ka];
      b0.y = B[(size_t)ncol0 * K + ka + 1];
      b1.x = B[(size_t)ncol1 * K + ka];
      b1.y = B[(size_t)ncol1 * K + ka + 1];
    } else {                                 // b[v] = B[ka+v][ncol]
      b0.x = B[(size_t)ka * N + ncol0];
      b0.y = B[(size_t)(ka + 1) * N + ncol0];
      b1.x = B[(size_t)ka * N + ncol1];
      b1.y = B[(size_t)(ka + 1) * N + ncol1];
    }
    acc0 = __builtin_amdgcn_wmma_f32_16x16x4_f32(false, a, false, b0,
                                                 (short)0, acc0, false, false);
    acc1 = __builtin_amdgcn_wmma_f32_16x16x4_f32(false, a, false, b1,
                                                 (short)0, acc1, false, false);
  }

  const int crow = m0 + ((lane >> 4) << 3);  // D: vgpr v -> row m0+v (+8 hi lanes)
  float badd0 = 0.0f, badd1 = 0.0f;
  if constexpr (BIAS) { badd0 = bias[ncol0]; badd1 = bias[ncol1]; }
#pragma unroll
  for (int v = 0; v < 8; ++v) {
    C[(size_t)(crow + v) * N + ncol0] = acc0[v] + badd0;
    C[(size_t)(crow + v) * N + ncol1] = acc1[v] + badd1;
  }
}

// ------------------------------------------------------------------
// Degree by target node (float counts, atomic)
// ------------------------------------------------------------------
__global__ void k_deg(const int* __restrict__ col, int E, float* __restrict__ deg) {
  int i = blockIdx.x * blockDim.x + threadIdx.x;
  int st = gridDim.x * blockDim.x;
  for (; i < E; i += st) atomicAdd(&deg[col[i]], 1.0f);
}

// dinv = rsqrt(deg + 1)   (self-loop folded in; deg >= 1 always)
__global__ void k_dinv(const float* __restrict__ deg, float* __restrict__ dinv, int n) {
  int i = blockIdx.x * blockDim.x + threadIdx.x;
  if (i < n) dinv[i] = rsqrtf(deg[i] + 1.0f);
}

// ------------------------------------------------------------------
// Edge aggregation: agg[col] += dinv[row]*dinv[col] * xw[row]
// one edge per block-iteration, 256 channel threads (coalesced xw row)
// ------------------------------------------------------------------
__global__ void k_agg(const int* __restrict__ row, const int* __restrict__ col,
                      const float* __restrict__ dinv, const float* __restrict__ xw,
                      float* __restrict__ agg, int E) {
  int t = threadIdx.x;
  for (int e = blockIdx.x; e < E; e += gridDim.x) {
    int r = row[e], c = col[e];
    float norm = dinv[r] * dinv[c];
    atomicAdd(&agg[(size_t)c * HDIM + t], norm * xw[(size_t)r * HDIM + t]);
  }
}

// self-loop term + bias + ReLU (in place: agg -> hgcn)
__global__ void k_gcn_finish(float* __restrict__ agg, const float* __restrict__ xw,
                             const float* __restrict__ dinv, const float* __restrict__ gcn_b,
                             int Nn) {
  long total = (long)Nn * HDIM;
  long i = (long)blockIdx.x * blockDim.x + threadIdx.x;
  long st = (long)gridDim.x * blockDim.x;
  for (; i < total; i += st) {
    int node = (int)(i >> 8);      // /HDIM
    int c    = (int)(i & 255);
    float di = dinv[node];
    float v  = agg[i] + di * di * xw[i] + gcn_b[c];
    agg[i] = v > 0.0f ? v : 0.0f;
  }
}

// ------------------------------------------------------------------
// Sequential GRU scan: single 768-thread workgroup (24 wave32 on one WGP).
// h lives in LDS; thread j owns gate row j of w_hh, streamed from L2 as
// float4 (global_load_b128); h read as float4 (ds_load_b128). Prefetch
// next step's x_proj line (global_prefetch_b8) during the dot product.
// ------------------------------------------------------------------
__device__ __forceinline__ float sigmf(float x) { return 1.0f / (1.0f + __expf(-x)); }

__global__ void __launch_bounds__(G3)
k_gru(const float* __restrict__ xp, const float* __restrict__ w_hh,
      const float* __restrict__ b_hh, const float* __restrict__ h0,
      float* __restrict__ node_emb, float* __restrict__ hidden_out, int T) {
  __shared__ float h[HDIM];
  __shared__ float rz[2 * HDIM];
  const int j = threadIdx.x;
  if (j < HDIM) h[j] = h0[j];
  __syncthreads();

  const float bj = b_hh[j];
  const float4* __restrict__ wrow4 =
      reinterpret_cast<const float4*>(w_hh + (size_t)j * HDIM);
  const float4* h4 = reinterpret_cast<const float4*>(h);

  for (int t = 0; t < T; ++t) {
    if (t + 1 < T)
      __builtin_prefetch(&xp[(size_t)(t + 1) * G3 + j], 0, 3);
    float gh = bj;
#pragma unroll 8
    for (int k = 0; k < HDIM / 4; ++k) {
      float4 hv = h4[k];
      float4 wv = wrow4[k];
      gh = fmaf(hv.x, wv.x, gh);
      gh = fmaf(hv.y, wv.y, gh);
      gh = fmaf(hv.z, wv.z, gh);
      gh = fmaf(hv.w, wv.w, gh);
    }
    float xv = xp[(size_t)t * G3 + j];
    __syncthreads();                            // all reads of h done
    if (j < 2 * HDIM) rz[j] = sigmf(xv + gh);   // r (0..255), z (256..511)
    __syncthreads();
    if (j >= 2 * HDIM) {
      int jj = j - 2 * HDIM;
      float nv = tanhf(xv + rz[jj] * gh);
      float z  = rz[HDIM + jj];
      float hn = (1.0f - z) * nv + z * h[jj];
      h[jj] = hn;
      node_emb[(size_t)t * HDIM + jj] = hn;
    }
    __syncthreads();                            // publish new h
  }
  if (j < HDIM) hidden_out[j] = h[j];
}

// ------------------------------------------------------------------
// Edge readout: one wave32 per edge; float4 gathers (b128) + wave32
// shuffle reduction.  out[e] = sum_c 0.5*(ne[s][c]+ne[t][c])*w[c] + b
// ------------------------------------------------------------------
__global__ void k_edge_out(const int* __restrict__ src, const int* __restrict__ tgt,
                           const float* __restrict__ ne, const float* __restrict__ mlp_w,
                           const float* __restrict__ mlp_b, float* __restrict__ out, int E) {
  int wid  = (int)((blockIdx.x * blockDim.x + threadIdx.x) >> 5);
  int lane = threadIdx.x & 31;
  int nw   = (gridDim.x * blockDim.x) >> 5;
  const float4* __restrict__ w4 = reinterpret_cast<const float4*>(mlp_w);
  for (int e = wid; e < E; e += nw) {
    int s = src[e], t = tgt[e];
    const float4* ps4 = reinterpret_cast<const float4*>(ne + (size_t)s * HDIM);
    const float4* pt4 = reinterpret_cast<const float4*>(ne + (size_t)t * HDIM);
    float acc = 0.0f;
#pragma unroll
    for (int b = 0; b < HDIM / 4; b += 32) {
      float4 av = ps4[b + lane];
      float4 bv = pt4[b + lane];
      float4 wv = w4[b + lane];
      acc = fmaf(0.5f * (av.x + bv.x), wv.x, acc);
      acc = fmaf(0.5f * (av.y + bv.y), wv.y, acc);
      acc = fmaf(0.5f * (av.z + bv.z), wv.z, acc);
      acc = fmaf(0.5f * (av.w + bv.w), wv.w, acc);
    }
#pragma unroll
    for (int off = 16; off > 0; off >>= 1)
      acc += __shfl_down(acc, off, 32);
    if (lane == 0) out[e] = acc + mlp_b[0];
  }
}

// ------------------------------------------------------------------
extern "C" void kernel_launch(void* const* d_in, const int* in_sizes, int n_in,
                              void* d_out, int out_size, void* d_ws, size_t ws_size,
                              hipStream_t stream) {
  const float* x      = (const float*)d_in[0];
  const int*   ei     = (const int*)d_in[1];
  const float* hidden = (const float*)d_in[2];
  const float* gcn_w  = (const float*)d_in[3];
  const float* gcn_b  = (const float*)d_in[4];
  const float* w_ih   = (const float*)d_in[5];
  const float* w_hh   = (const float*)d_in[6];
  const float* b_ih   = (const float*)d_in[7];
  const float* b_hh   = (const float*)d_in[8];
  const float* mlp_w  = (const float*)d_in[9];
  const float* mlp_b  = (const float*)d_in[10];

  const int Nn = in_sizes[0] / IN_C;   // 10000
  const int E  = in_sizes[1] / 2;      // 320000
  const int* row = ei;                 // edge_index[0] = src
  const int* col = ei + E;             // edge_index[1] = tgt

  float* ws    = (float*)d_ws;
  float* xw    = ws;                               // Nn*H
  float* agg   = xw   + (size_t)Nn * HDIM;         // Nn*H   (becomes hgcn in place)
  float* deg   = agg  + (size_t)Nn * HDIM;         // Nn
  float* dinv  = deg  + Nn;                        // Nn
  float* xproj = dinv + Nn;                        // Nn*G3
  float* nemb  = xproj + (size_t)Nn * G3;          // Nn*H

  float* out = (float*)d_out;          // [0..E) edge_out, [E..E+H) hidden_out

  // 1. zero agg + deg (contiguous region)
  k_zero<<<2048, 256, 0, stream>>>(agg, (long)Nn * HDIM + Nn);

  // 2. xw = x @ gcn_w          (10000x128 @ 128x256), B row-major
  {
    int waves = (Nn / 16) * (HDIM / 32);           // 5000 waves (16x32 strips)
    k_gemm_wmma<false, false><<<(waves * 32 + 255) / 256, 256, 0, stream>>>(
        x, gcn_w, nullptr, xw, Nn, HDIM, IN_C);
  }

  // 3-4. degree -> dinv
  k_deg<<<1024, 256, 0, stream>>>(col, E, deg);
  k_dinv<<<(Nn + 255) / 256, 256, 0, stream>>>(deg, dinv, Nn);

  // 5. normalized scatter-add aggregation
  k_agg<<<4096, HDIM, 0, stream>>>(row, col, dinv, xw, agg, E);

  // 6. self-loop + bias + ReLU  (agg -> hgcn in place)
  k_gcn_finish<<<4096, 256, 0, stream>>>(agg, xw, dinv, gcn_b, Nn);

  // 7. x_proj = hgcn @ w_ih^T + b_ih   (10000x256 @ 256x768), B transposed
  {
    int waves = (Nn / 16) * (G3 / 32);             // 15000 waves
    k_gemm_wmma<true, true><<<(waves * 32 + 255) / 256, 256, 0, stream>>>(
        agg, w_ih, b_ih, xproj, Nn, G3, HDIM);
  }

  // 8. sequential GRU scan (one workgroup), writes node_emb + hidden_out
  k_gru<<<1, G3, 0, stream>>>(xproj, w_hh, b_hh, hidden, nemb, out + E, Nn);

  // 9. edge readout
  k_edge_out<<<4096, 256, 0, stream>>>(row, col, nemb, mlp_w, mlp_b, out, E);
}